// HierarchicalSoftmax_18133351924188
// MI455X (gfx1250) — compile-verified
//
#include <hip/hip_runtime.h>
#include <hip/hip_bf16.h>

// Problem constants (from the reference)
#define NHID   128
#define NCLS   320     // nclasses == ntok_per_class == 320
#define BATCH  1024
#define NTOK   (NCLS * NCLS)   // 102400 output columns per row

typedef float v2f __attribute__((ext_vector_type(2)));
typedef float v4f __attribute__((ext_vector_type(4)));
typedef float v8f __attribute__((ext_vector_type(8)));

// ---------------------------------------------------------------------------
// Kernel 1: WMMA f32 GEMMs.
//   TOP[b][n] = sum_k X[b][k] * Wt[k][n] + bt[n]      (X @ Wt)
//   BOT[b][n] = sum_k X[b][k] * Wb[n][k] + bb[n]      (X @ Wb^T)
// One wave per 16x16 tile of both outputs; K-loop in steps of 4 using
// V_WMMA_F32_16X16X4_F32 (exact f32 path -> matches f32 reference).
//
// VGPR layouts (ISA 7.12.2, 32-bit, wave32):
//   A 16x4 : lane = grp*16+idx ; a[d] = A[M=idx][k0 + 2*grp + d]
//   B 4x16 : lane = grp*16+idx ; b[d] = B[K=k0 + 2*grp + d][N=idx]
//   C 16x16: c[r] at lane = C[M = r + 8*grp][N = idx]
// ---------------------------------------------------------------------------
__global__ __launch_bounds__(128)
void hsm_gemm_wmma(const float* __restrict__ X,    // [BATCH, NHID]
                   const float* __restrict__ Wt,   // [NHID, NCLS] row-major
                   const float* __restrict__ bt,   // [NCLS]
                   const float* __restrict__ Wb,   // [NCLS, NHID] row-major
                   const float* __restrict__ bb,   // [NCLS]
                   float* __restrict__ TOP,        // [BATCH, NCLS]
                   float* __restrict__ BOT)        // [BATCH, NCLS]
{
    const int wave = (blockIdx.x * blockDim.x + threadIdx.x) >> 5;
    const int lane = threadIdx.x & 31;
    const int mt = wave / (NCLS / 16);   // 0..63
    const int nt = wave % (NCLS / 16);   // 0..19
    const int grp = lane >> 4;           // 0/1 : selects K pair
    const int idx = lane & 15;           // row (A) / col (B,C)
    const int m0 = mt * 16;
    const int n0 = nt * 16;

    // Per-lane base pointers (include the 2*grp K shift).
    const float* arow  = X  + (size_t)(m0 + idx) * NHID + 2 * grp;           // A[M=idx]
    const float* btcol = Wt + (size_t)(2 * grp) * NCLS + (n0 + idx);         // Wt[k][n]
    const float* bbrow = Wb + (size_t)(n0 + idx) * NHID + 2 * grp;           // Wb[n][k]

    v8f ct = {};
    v8f cb = {};

    #pragma unroll 4
    for (int k0 = 0; k0 < NHID; k0 += 4) {
        // A: contiguous pair -> single b64 load
        v2f a = *(const v2f*)(arow + k0);
        // B (top): stride-NCLS in k -> two b32 loads
        v2f btv;
        btv.x = btcol[(size_t)k0 * NCLS];
        btv.y = btcol[(size_t)(k0 + 1) * NCLS];
        // B (bottom): Wb^T -> contiguous in k -> single b64 load
        v2f bbv = *(const v2f*)(bbrow + k0);

        ct = __builtin_amdgcn_wmma_f32_16x16x4_f32(
                 false, a, false, btv, (short)0, ct, false, false);
        cb = __builtin_amdgcn_wmma_f32_16x16x4_f32(
                 false, a, false, bbv, (short)0, cb, false, false);
    }

    const float biast = bt[n0 + idx];
    const float biasb = bb[n0 + idx];

    #pragma unroll
    for (int r = 0; r < 8; ++r) {
        const int m = m0 + r + 8 * grp;
        TOP[(size_t)m * NCLS + n0 + idx] = ct[r] + biast;
        BOT[(size_t)m * NCLS + n0 + idx] = cb[r] + biasb;
    }
}

// ---------------------------------------------------------------------------
// Kernel 2: broadcast outer-add, store-bandwidth bound (419 MB of output).
//   out[b][c*320 + t] = TOP[b][c] + BOT[b][t]
// One block per batch row. Stage the two 320-float rows in LDS, then stream
// float4 non-temporal stores (output is 2x L2 capacity -> don't cache it).
// 320 % 4 == 0, so a float4 never crosses a class boundary: constant c per
// vector, aligned b128 read of BOT[t..t+3] from LDS.
// ---------------------------------------------------------------------------
__global__ __launch_bounds__(256)
void hsm_bcast_add(const float* __restrict__ TOP,
                   const float* __restrict__ BOT,
                   float* __restrict__ out)
{
    __shared__ float s_top[NCLS];
    __shared__ float s_bot[NCLS];

    const int b   = blockIdx.x;
    const int tid = threadIdx.x;

    for (int i = tid; i < NCLS; i += 256) {
        s_top[i] = TOP[(size_t)b * NCLS + i];
        s_bot[i] = BOT[(size_t)b * NCLS + i];
    }
    __syncthreads();

    const size_t rowbase = (size_t)b * NTOK;
    const int nvec = NTOK / 4;  // 25600 float4 per row

    for (int i = tid; i < nvec; i += 256) {
        const int col = i * 4;
        const int c   = col / NCLS;          // class index (const within vec)
        const int t   = col - c * NCLS;      // token-in-class base (mult of 4)
        const float tv = s_top[c];
        const v4f bv = *(const v4f*)&s_bot[t];
        v4f o;
        o.x = tv + bv.x;
        o.y = tv + bv.y;
        o.z = tv + bv.z;
        o.w = tv + bv.w;
        __builtin_nontemporal_store(o, (v4f*)(out + rowbase + col));
    }
}

// ---------------------------------------------------------------------------
extern "C" void kernel_launch(void* const* d_in, const int* in_sizes, int n_in,
                              void* d_out, int out_size, void* d_ws, size_t ws_size,
                              hipStream_t stream)
{
    const float* X  = (const float*)d_in[0];  // [1024,128]
    const float* Wt = (const float*)d_in[1];  // [128,320]
    const float* bt = (const float*)d_in[2];  // [320]
    const float* Wb = (const float*)d_in[3];  // [320,128]
    const float* bb = (const float*)d_in[4];  // [320]
    float* out = (float*)d_out;               // [1024,102400]

    float* TOP = (float*)d_ws;                // [1024,320]
    float* BOT = TOP + (size_t)BATCH * NCLS;  // [1024,320]  (2.6 MB total)

    // 1280 tiles (64 x 20), one wave each; 4 waves per 128-thread block.
    const int nwaves  = (BATCH / 16) * (NCLS / 16);   // 1280
    const int nblocks = nwaves / 4;                   // 320
    hsm_gemm_wmma<<<nblocks, 128, 0, stream>>>(X, Wt, bt, Wb, bb, TOP, BOT);

    // One block per batch row; 256 threads x 100 float4 NT stores each.
    hsm_bcast_add<<<BATCH, 256, 0, stream>>>(TOP, BOT, out);
}